// LogicGNN_68917045232210
// MI455X (gfx1250) — compile-verified
//
#include <hip/hip_runtime.h>
#include <math.h>

typedef float v2f __attribute__((ext_vector_type(2)));
typedef float v8f __attribute__((ext_vector_type(8)));

#define GAT_NEG_SLOPE 0.2f

// ---------------------------------------------------------------------------
// utility kernels
// ---------------------------------------------------------------------------
__global__ void gat_fill_kernel(float* __restrict__ p, float v, int n) {
    int i = blockIdx.x * blockDim.x + threadIdx.x;
    if (i < n) p[i] = v;
}

__global__ void gat_recip_kernel(float* __restrict__ p, int n) {
    int i = blockIdx.x * blockDim.x + threadIdx.x;
    if (i < n) p[i] = 1.0f / p[i];
}

// ---------------------------------------------------------------------------
// fp32 WMMA GEMM: C[M,Nn] = A[M,K] @ B[K,Nn], row-major.
// Requires M%16==0, K%4==0, Nn%64==0 (true here: M=50000=16*3125, K=128/256,
// Nn=256/128). One wave computes a 16x64 output strip using 4 accumulators
// chained over K in steps of 4 via V_WMMA_F32_16X16X4_F32.
//
// A 16x4 layout : lane t holds A[m=t&15, k0+2*(t>>4) + {0,1}]   (v2f)
// B 4x16 layout : lane t holds B[k0+2*(t>>4) + {0,1}, n=t&15]   (v2f)
// D 16x16 layout: lane t, vgpr r holds D[r + 8*(t>>4), t&15]    (v8f)
// ---------------------------------------------------------------------------
__global__ void gat_wmma_gemm_f32(const float* __restrict__ A,
                                  const float* __restrict__ B,
                                  float* __restrict__ C,
                                  int M, int K, int Nn) {
    const int wavesPerBlock = blockDim.x >> 5;
    const int w = blockIdx.x * wavesPerBlock + (threadIdx.x >> 5);
    const int tilesM = M >> 4;
    const int tilesN = Nn >> 6;
    if (w >= tilesM * tilesN) return;          // wave-uniform: EXEC stays all-1

    const int tm = w % tilesM;
    const int tn = w / tilesM;
    const int lane = threadIdx.x & 31;
    const int rc = lane & 15;                  // A-row / B-col within tile
    const int kh = lane >> 4;                  // selects K pair {0,1} vs {2,3}
    const int m0 = tm << 4;
    const int n0 = tn << 6;

    v8f acc0 = {0}, acc1 = {0}, acc2 = {0}, acc3 = {0};

    const float* arow = A + (size_t)(m0 + rc) * K + 2 * kh;
    for (int k = 0; k < K; k += 4) {
        v2f a = *(const v2f*)(arow + k);       // 8B-aligned contiguous pair
        const float* bbase = B + (size_t)(k + 2 * kh) * Nn + n0 + rc;
        v2f b0, b1, b2, b3;
        b0.x = bbase[0];  b0.y = bbase[Nn + 0];
        b1.x = bbase[16]; b1.y = bbase[Nn + 16];
        b2.x = bbase[32]; b2.y = bbase[Nn + 32];
        b3.x = bbase[48]; b3.y = bbase[Nn + 48];
        acc0 = __builtin_amdgcn_wmma_f32_16x16x4_f32(false, a, false, b0, (short)0, acc0, false, false);
        acc1 = __builtin_amdgcn_wmma_f32_16x16x4_f32(false, a, false, b1, (short)0, acc1, false, false);
        acc2 = __builtin_amdgcn_wmma_f32_16x16x4_f32(false, a, false, b2, (short)0, acc2, false, false);
        acc3 = __builtin_amdgcn_wmma_f32_16x16x4_f32(false, a, false, b3, (short)0, acc3, false, false);
    }

    float* crow = C + (size_t)(m0 + 8 * kh) * Nn + n0 + rc;
#pragma unroll
    for (int r = 0; r < 8; r++) {
        crow[(size_t)r * Nn + 0]  = acc0[r];
        crow[(size_t)r * Nn + 16] = acc1[r];
        crow[(size_t)r * Nn + 32] = acc2[r];
        crow[(size_t)r * Nn + 48] = acc3[r];
    }
}

// ---------------------------------------------------------------------------
// per-(node, head) attention dot products: as[n,h] = <h[n,h,:], att_src[h,:]>
// ---------------------------------------------------------------------------
__global__ void gat_attn_dot_kernel(const float* __restrict__ h,
                                    const float* __restrict__ att_s,
                                    const float* __restrict__ att_d,
                                    float* __restrict__ as_,
                                    float* __restrict__ ad_,
                                    int nNodes, int H, int C) {
    int tid = blockIdx.x * blockDim.x + threadIdx.x;
    if (tid >= nNodes * H) return;
    int node = tid / H, head = tid % H;
    const float4* hp = (const float4*)(h + (size_t)node * H * C + (size_t)head * C);
    const float4* sp = (const float4*)(att_s + (size_t)head * C);
    const float4* dp = (const float4*)(att_d + (size_t)head * C);
    float s = 0.f, d = 0.f;
    int C4 = C >> 2;
    for (int i = 0; i < C4; i++) {
        float4 hv = hp[i], sv = sp[i], dv = dp[i];
        s += hv.x * sv.x + hv.y * sv.y + hv.z * sv.z + hv.w * sv.w;
        d += hv.x * dv.x + hv.y * dv.y + hv.z * dv.z + hv.w * dv.w;
    }
    as_[tid] = s;
    ad_[tid] = d;
}

// ---------------------------------------------------------------------------
// edge helpers
// ---------------------------------------------------------------------------
__device__ __forceinline__ void gat_edge_sd(const int* __restrict__ ei, int E,
                                            int e, int& s, int& d) {
    if (e < E) { s = ei[e]; d = ei[E + e]; }
    else       { s = e - E; d = s; }          // appended self-loops
}

__device__ __forceinline__ float gat_leaky(float x) {
    return x > 0.f ? x : GAT_NEG_SLOPE * x;
}

// float atomic max via sign-aware integer atomics (handles -0.0; no NaNs here)
__device__ __forceinline__ void gat_atomic_max_f32(float* addr, float val) {
    int iv = __float_as_int(val);
    if (iv >= 0) atomicMax((int*)addr, iv);
    else         atomicMin((unsigned int*)addr, (unsigned int)iv);
}

// pass 1: segment max over destination nodes
__global__ void gat_edge_max_kernel(const int* __restrict__ ei, int E, int E2,
                                    const float* __restrict__ as_,
                                    const float* __restrict__ ad_,
                                    float* __restrict__ m, int H) {
    int tid = blockIdx.x * blockDim.x + threadIdx.x;
    if (tid >= E2 * H) return;
    int e = tid / H, hh = tid - e * H;
    int s, d; gat_edge_sd(ei, E, e, s, d);
    float ev = gat_leaky(as_[(size_t)s * H + hh] + ad_[(size_t)d * H + hh]);
    gat_atomic_max_f32(&m[(size_t)d * H + hh], ev);
}

// pass 2: ex = exp(e - m[dst]); accumulate denom; stash ex per (edge, head)
__global__ void gat_edge_sum_kernel(const int* __restrict__ ei, int E, int E2,
                                    const float* __restrict__ as_,
                                    const float* __restrict__ ad_,
                                    const float* __restrict__ m,
                                    float* __restrict__ den,
                                    float* __restrict__ exw, int H) {
    int tid = blockIdx.x * blockDim.x + threadIdx.x;
    if (tid >= E2 * H) return;
    int e = tid / H, hh = tid - e * H;
    int s, d; gat_edge_sd(ei, E, e, s, d);
    float ev = gat_leaky(as_[(size_t)s * H + hh] + ad_[(size_t)d * H + hh]);
    float ex = __expf(ev - m[(size_t)d * H + hh]);
    exw[tid] = ex;
    atomicAdd(&den[(size_t)d * H + hh], ex);
}

// pass 3: agg[dst,:] += alpha * h[src,:]  — one wave32 per edge, coalesced,
// f32 global atomics resolve in the 192MB L2 (feature arrays are L2-resident)
__global__ void gat_edge_agg_kernel(const int* __restrict__ ei, int E, int E2,
                                    const float* __restrict__ exw,
                                    const float* __restrict__ rden,
                                    const float* __restrict__ h,
                                    float* __restrict__ agg, int H, int C) {
    int w = blockIdx.x * (blockDim.x >> 5) + (threadIdx.x >> 5);
    if (w >= E2) return;
    int lane = threadIdx.x & 31;
    int s, d; gat_edge_sd(ei, E, w, s, d);
    int HC = H * C;
    const float* hs = h + (size_t)s * HC;
    float* ag = agg + (size_t)d * HC;
    for (int i = lane; i < HC; i += 32) {
        int hh = i / C;
        float alpha = exw[(size_t)w * H + hh] * rden[(size_t)d * H + hh];
        atomicAdd(&ag[i], alpha * hs[i]);
    }
}

// y = elu(y + b) in place
__global__ void gat_bias_elu_kernel(float* __restrict__ y,
                                    const float* __restrict__ b,
                                    int HC, int total) {
    int i = blockIdx.x * blockDim.x + threadIdx.x;
    if (i >= total) return;
    float v = y[i] + b[i % HC];
    y[i] = v > 0.f ? v : (__expf(v) - 1.f);
}

// y += b in place
__global__ void gat_bias_add_kernel(float* __restrict__ y,
                                    const float* __restrict__ b,
                                    int HC, int total) {
    int i = blockIdx.x * blockDim.x + threadIdx.x;
    if (i >= total) return;
    y[i] += b[i % HC];
}

// ---------------------------------------------------------------------------
// launch
// ---------------------------------------------------------------------------
extern "C" void kernel_launch(void* const* d_in, const int* in_sizes, int n_in,
                              void* d_out, int out_size, void* d_ws, size_t ws_size,
                              hipStream_t stream) {
    const float* x   = (const float*)d_in[0];
    const int*   ei  = (const int*)d_in[1];
    const float* W1  = (const float*)d_in[2];
    const float* aS1 = (const float*)d_in[3];
    const float* aD1 = (const float*)d_in[4];
    const float* b1  = (const float*)d_in[5];
    const float* W2  = (const float*)d_in[6];
    const float* aS2 = (const float*)d_in[7];
    const float* aD2 = (const float*)d_in[8];
    const float* b2  = (const float*)d_in[9];
    float* out = (float*)d_out;

    const int IN_F = 128, HID = 64, OUTC = 128, H1 = 4;
    const int HC1 = H1 * HID;                 // 256
    const int nNodes = in_sizes[0] / IN_F;    // 50000
    const int E  = in_sizes[1] / 2;           // 800000
    const int E2 = E + nNodes;                // + self loops

    float* ws = (float*)d_ws;
    size_t off = 0;
    float* h1   = ws + off; off += (size_t)nNodes * HC1;   // pre-attn features L1
    float* agg1 = ws + off; off += (size_t)nNodes * HC1;   // L1 aggregate -> y (in place)
    float* as1  = ws + off; off += (size_t)nNodes * H1;
    float* ad1  = ws + off; off += (size_t)nNodes * H1;
    float* m1   = ws + off; off += (size_t)nNodes * H1;
    float* den1 = ws + off; off += (size_t)nNodes * H1;
    float* ex1  = ws + off; off += (size_t)E2 * H1;
    float* h2   = ws + off; off += (size_t)nNodes * OUTC;
    float* as2  = ws + off; off += (size_t)nNodes;
    float* ad2  = ws + off; off += (size_t)nNodes;
    float* m2   = ws + off; off += (size_t)nNodes;
    float* den2 = ws + off; off += (size_t)nNodes;
    float* ex2  = ws + off; off += (size_t)E2;
    (void)ws_size; (void)n_in; (void)out_size;

    const int TB = 256;
    auto cdiv = [](long long a, long long b) { return (int)((a + b - 1) / b); };

    // ===================== layer 1 =====================
    gat_wmma_gemm_f32<<<cdiv((long long)(nNodes / 16) * (HC1 / 64), 8), TB, 0, stream>>>(
        x, W1, h1, nNodes, IN_F, HC1);
    gat_attn_dot_kernel<<<cdiv((long long)nNodes * H1, TB), TB, 0, stream>>>(
        h1, aS1, aD1, as1, ad1, nNodes, H1, HID);
    gat_fill_kernel<<<cdiv((long long)nNodes * H1, TB), TB, 0, stream>>>(m1, -INFINITY, nNodes * H1);
    gat_fill_kernel<<<cdiv((long long)nNodes * H1, TB), TB, 0, stream>>>(den1, 0.f, nNodes * H1);
    gat_fill_kernel<<<cdiv((long long)nNodes * HC1, TB), TB, 0, stream>>>(agg1, 0.f, nNodes * HC1);
    gat_edge_max_kernel<<<cdiv((long long)E2 * H1, TB), TB, 0, stream>>>(ei, E, E2, as1, ad1, m1, H1);
    gat_edge_sum_kernel<<<cdiv((long long)E2 * H1, TB), TB, 0, stream>>>(ei, E, E2, as1, ad1, m1, den1, ex1, H1);
    gat_recip_kernel<<<cdiv((long long)nNodes * H1, TB), TB, 0, stream>>>(den1, nNodes * H1);
    gat_edge_agg_kernel<<<cdiv(E2, 8), TB, 0, stream>>>(ei, E, E2, ex1, den1, h1, agg1, H1, HID);
    gat_bias_elu_kernel<<<cdiv((long long)nNodes * HC1, TB), TB, 0, stream>>>(agg1, b1, HC1, nNodes * HC1);

    // ===================== layer 2 =====================
    gat_wmma_gemm_f32<<<cdiv((long long)(nNodes / 16) * (OUTC / 64), 8), TB, 0, stream>>>(
        agg1, W2, h2, nNodes, HC1, OUTC);
    gat_attn_dot_kernel<<<cdiv((long long)nNodes, TB), TB, 0, stream>>>(
        h2, aS2, aD2, as2, ad2, nNodes, 1, OUTC);
    gat_fill_kernel<<<cdiv((long long)nNodes, TB), TB, 0, stream>>>(m2, -INFINITY, nNodes);
    gat_fill_kernel<<<cdiv((long long)nNodes, TB), TB, 0, stream>>>(den2, 0.f, nNodes);
    gat_fill_kernel<<<cdiv((long long)nNodes * OUTC, TB), TB, 0, stream>>>(out, 0.f, nNodes * OUTC);
    gat_edge_max_kernel<<<cdiv((long long)E2, TB), TB, 0, stream>>>(ei, E, E2, as2, ad2, m2, 1);
    gat_edge_sum_kernel<<<cdiv((long long)E2, TB), TB, 0, stream>>>(ei, E, E2, as2, ad2, m2, den2, ex2, 1);
    gat_recip_kernel<<<cdiv((long long)nNodes, TB), TB, 0, stream>>>(den2, nNodes);
    gat_edge_agg_kernel<<<cdiv(E2, 8), TB, 0, stream>>>(ei, E, E2, ex2, den2, h2, out, 1, OUTC);
    gat_bias_add_kernel<<<cdiv((long long)nNodes * OUTC, TB), TB, 0, stream>>>(out, b2, OUTC, nNodes * OUTC);
}